// GraphClassificationModel_72894184948279
// MI455X (gfx1250) — compile-verified
//
#include <hip/hip_runtime.h>
#include <hip/hip_bf16.h>

// GIN graph classification on MI455X (gfx1250), restructured:
//   s[v]        = feat[v] + sum_{u->v} feat[u]                 (scalar edge pass)
//   h1[v][d]    = relu(s[v]*W1[d] + b1[d])                     (computed on the fly)
//   weight[v][g]= [graph(v)==g] + #{edges v->w : graph(w)==g}  (count-matrix edge pass)
//   pooled[g][d]= sum_v weight[v][g] * h1[v][d]                (WMMA f32 16x16x4 GEMM, K=100k)
//   hg          = (pooled @ W2)/cnt + b2 ; out = relu(hg@Wf1+bf1)@Wf2+bf2

typedef __attribute__((ext_vector_type(2))) float v2f;
typedef __attribute__((ext_vector_type(8))) float v8f;

#define EMBED   128
#define NGRAPH  128
#define NCLASS  10
#define KCHUNK  4000   // 100000 / 4000 = 25 K-splits, divisible by 8

// ---------------- stage 0: zero pooled + graph counts ----------------
__global__ void gin_zero_kernel(float* __restrict__ pooled, float* __restrict__ cnt) {
    int i = blockIdx.x * blockDim.x + threadIdx.x;
    if (i < NGRAPH * EMBED) pooled[i] = 0.0f;
    if (i < NGRAPH) cnt[i] = 0.0f;
}

// ---------------- stage 1: init s, weight matrix, graph counts ----------------
__global__ void gin_init_kernel(const float* __restrict__ feat, const int* __restrict__ gid,
                                float* __restrict__ s, float* __restrict__ Wt,
                                float* __restrict__ cnt, int n) {
    long long idx = (long long)blockIdx.x * blockDim.x + threadIdx.x;
    if (idx >= (long long)n * NGRAPH) return;
    int v = (int)(idx >> 7);
    int d = (int)(idx & (NGRAPH - 1));
    int g = gid[v];
    Wt[idx] = (d == g) ? 1.0f : 0.0f;          // the "[graph(v)==g]" term of weight
    if (d == 0) {
        s[v] = feat[v];                        // seed scalar aggregate with self term
        atomicAdd(&cnt[g], 1.0f);              // nodes per graph (for mean pooling)
    }
}

// ---------------- stage 2: edge pass (scalar aggregate + count matrix) ----------------
__global__ void gin_edge_kernel(const int* __restrict__ src, const int* __restrict__ dst,
                                const float* __restrict__ feat, const int* __restrict__ gid,
                                float* __restrict__ s, float* __restrict__ Wt, int nE) {
    int e = blockIdx.x * blockDim.x + threadIdx.x;
    if (e >= nE) return;
    int u = src[e];
    int v = dst[e];
    atomicAdd(&s[v], feat[u]);                          // scalar conv1 aggregation
    atomicAdd(&Wt[(long long)u * NGRAPH + gid[v]], 1.0f); // c[u][graph(v)] += 1
}

// ---------------- stage 3: pooled = Wt^T x relu(s*W1+b1), WMMA f32 16x16x4 ----------------
// grid = (64 tiles [8 graph-tiles x 8 dim-tiles], 25 K-splits), block = 1 wave32
__global__ __launch_bounds__(32)
void gin_pool_wmma_kernel(const float* __restrict__ Wt, const float* __restrict__ s,
                          const float* __restrict__ W1, const float* __restrict__ b1,
                          float* __restrict__ pooled) {
    const int lane = threadIdx.x;          // 0..31, wave32
    const int half = lane >> 4;            // 0: lanes 0-15, 1: lanes 16-31
    const int l15  = lane & 15;
    const int gb = (blockIdx.x >> 3) << 4; // graph tile base (M)
    const int db = (blockIdx.x & 7) << 4;  // embed tile base (N)
    const int k0 = blockIdx.y * KCHUNK;

    // per-lane column constants for on-the-fly h1 generation (B matrix column = db+l15)
    const float w  = W1[db + l15];
    const float bb = b1[db + l15];

    const float* __restrict__ WtA = Wt + gb + l15; // A[M=gb+l15][K] = Wt[K][gb+l15]

    v8f acc = {0.f, 0.f, 0.f, 0.f, 0.f, 0.f, 0.f, 0.f};

    // A 16x4 layout: v[0] = K0 (lanes 0-15) / K2 (lanes 16-31); v[1] = K1 / K3
    // B 4x16 layout: v[0] = row K0 (lanes 0-15) / row K2 (lanes 16-31); v[1] = K1 / K3
    for (int k = k0; k < k0 + KCHUNK; k += 8) {
        int ka = k + half * 2;                       // node for K-slot 0/2
        {
            v2f A; A.x = WtA[ka * NGRAPH]; A.y = WtA[(ka + 1) * NGRAPH];
            float sa = s[ka], sb = s[ka + 1];
            v2f B; B.x = fmaxf(fmaf(sa, w, bb), 0.f); B.y = fmaxf(fmaf(sb, w, bb), 0.f);
            acc = __builtin_amdgcn_wmma_f32_16x16x4_f32(false, A, false, B,
                                                        (short)0, acc, false, false);
        }
        {
            int kc = ka + 4;
            v2f A; A.x = WtA[kc * NGRAPH]; A.y = WtA[(kc + 1) * NGRAPH];
            float sc = s[kc], sd = s[kc + 1];
            v2f B; B.x = fmaxf(fmaf(sc, w, bb), 0.f); B.y = fmaxf(fmaf(sd, w, bb), 0.f);
            acc = __builtin_amdgcn_wmma_f32_16x16x4_f32(false, A, false, B,
                                                        (short)0, acc, false, false);
        }
    }

    // D 16x16 f32 layout: vgpr j -> row gb+j (lanes 0-15) / row gb+j+8 (lanes 16-31)
    #pragma unroll
    for (int j = 0; j < 8; ++j) {
        int g = gb + j + half * 8;
        atomicAdd(&pooled[g * EMBED + db + l15], acc[j]);
    }
}

// ---------------- stage 4: hg = (pooled @ W2)/max(cnt,1) + b2 ----------------
__global__ void gin_fc1_kernel(const float* __restrict__ pooled, const float* __restrict__ cnt,
                               const float* __restrict__ W2, const float* __restrict__ b2,
                               float* __restrict__ hg) {
    int idx = blockIdx.x * blockDim.x + threadIdx.x;
    if (idx >= NGRAPH * EMBED) return;
    int g = idx >> 7, d = idx & (EMBED - 1);
    float inv = 1.0f / fmaxf(cnt[g], 1.0f);
    float acc = 0.0f;
    for (int k = 0; k < EMBED; ++k)
        acc = fmaf(pooled[g * EMBED + k], W2[k * EMBED + d], acc);
    hg[idx] = acc * inv + b2[d];
}

// ---------------- stage 5: t = relu(hg @ Wf1 + bf1) ----------------
__global__ void gin_fc2_kernel(const float* __restrict__ hg, const float* __restrict__ Wf1,
                               const float* __restrict__ bf1, float* __restrict__ t) {
    int idx = blockIdx.x * blockDim.x + threadIdx.x;
    if (idx >= NGRAPH * EMBED) return;
    int g = idx >> 7, d = idx & (EMBED - 1);
    float acc = bf1[d];
    for (int k = 0; k < EMBED; ++k)
        acc = fmaf(hg[g * EMBED + k], Wf1[k * EMBED + d], acc);
    t[idx] = fmaxf(acc, 0.0f);
}

// ---------------- stage 6: out = t @ Wf2 + bf2 ----------------
__global__ void gin_fc3_kernel(const float* __restrict__ t, const float* __restrict__ Wf2,
                               const float* __restrict__ bf2, float* __restrict__ out) {
    int idx = blockIdx.x * blockDim.x + threadIdx.x;
    if (idx >= NGRAPH * NCLASS) return;
    int g = idx / NCLASS, c = idx % NCLASS;
    float acc = bf2[c];
    for (int k = 0; k < EMBED; ++k)
        acc = fmaf(t[g * EMBED + k], Wf2[k * NCLASS + c], acc);
    out[idx] = acc;
}

extern "C" void kernel_launch(void* const* d_in, const int* in_sizes, int n_in,
                              void* d_out, int out_size, void* d_ws, size_t ws_size,
                              hipStream_t stream) {
    const float* feat = (const float*)d_in[0];
    const int*   src  = (const int*)d_in[1];
    const int*   dst  = (const int*)d_in[2];
    const int*   gid  = (const int*)d_in[3];
    const float* W1   = (const float*)d_in[4];
    const float* b1   = (const float*)d_in[5];
    const float* W2   = (const float*)d_in[6];
    const float* b2   = (const float*)d_in[7];
    const float* Wf1  = (const float*)d_in[8];
    const float* bf1  = (const float*)d_in[9];
    const float* Wf2  = (const float*)d_in[10];
    const float* bf2  = (const float*)d_in[11];
    float* out = (float*)d_out;

    const int N  = in_sizes[0];   // 100000
    const int nE = in_sizes[1];   // 3200000

    // workspace carve-up (bytes, 16B aligned)
    char* ws = (char*)d_ws;
    float* s      = (float*)(ws);                                 // N floats
    float* Wt     = (float*)(ws + (size_t)400000);                // N*128 floats
    float* pooled = (float*)(ws + (size_t)400000 + 51200000);     // 128*128
    float* cnt    = (float*)(ws + (size_t)400000 + 51200000 + 65536);
    float* hg     = (float*)(ws + (size_t)400000 + 51200000 + 65536 + 512);
    float* t1     = (float*)(ws + (size_t)400000 + 51200000 + 65536 + 512 + 65536);

    // stage 0: zero accumulators
    gin_zero_kernel<<<(NGRAPH * EMBED + 255) / 256, 256, 0, stream>>>(pooled, cnt);

    // stage 1: init s = feat, Wt[v][g] = [graph(v)==g], cnt[g] = node count
    {
        long long total = (long long)N * NGRAPH;
        int blocks = (int)((total + 255) / 256);
        gin_init_kernel<<<blocks, 256, 0, stream>>>(feat, gid, s, Wt, cnt, N);
    }

    // stage 2: edge pass
    gin_edge_kernel<<<(nE + 255) / 256, 256, 0, stream>>>(src, dst, feat, gid, s, Wt, nE);

    // stage 3: WMMA pooled GEMM, K split across gridDim.y
    {
        dim3 grid(64, N / KCHUNK);  // N = 100000 -> 25 K-splits
        gin_pool_wmma_kernel<<<grid, 32, 0, stream>>>(Wt, s, W1, b1, pooled);
    }

    // stages 4-6: small FC chain
    gin_fc1_kernel<<<(NGRAPH * EMBED + 255) / 256, 256, 0, stream>>>(pooled, cnt, W2, b2, hg);
    gin_fc2_kernel<<<(NGRAPH * EMBED + 255) / 256, 256, 0, stream>>>(hg, Wf1, bf1, t1);
    gin_fc3_kernel<<<(NGRAPH * NCLASS + 255) / 256, 256, 0, stream>>>(t1, Wf2, bf2, out);
}